// Gate_90426241450822
// MI455X (gfx1250) — compile-verified
//
#include <hip/hip_runtime.h>
#include <math.h>

// ---------------------------------------------------------------------------
// MoE gate: logits = x @ W  ([16384,4096] x [4096,64]), softmax + top-2 mask.
// Bandwidth-bound: 256 MB of x @ 23.3 TB/s ~= 11 us floor. fp32 WMMA
// (v_wmma_f32_16x16x4_f32) keeps full reference precision for the logits
// output. Data movement: double-buffered async global->LDS copies (ASYNCcnt).
// W is stored pair-interleaved in LDS so each B fragment is one ds_load_b64
// into an even-aligned VGPR pair (no repacking movs).
// ---------------------------------------------------------------------------

typedef __attribute__((ext_vector_type(2))) float v2f;
typedef __attribute__((ext_vector_type(8))) float v8f;
typedef __attribute__((ext_vector_type(4))) int   v4i;

#define AS1 __attribute__((address_space(1)))
#define AS3 __attribute__((address_space(3)))

#define D_MODEL   4096
#define EXPERTS   64
#define TOKENS    (4 * 4096)        // B*S = 16384
#define BK        32                // K chunk staged through LDS (double-buffered)
#define XS_STRIDE (BK + 4)          // 36: A-frag ds_load_b64 conflict-free (gcd trick)
#define WP_STRIDE 160               // pair-row stride (2*64 data + 32 pad dwords)
                                    // 160 % 64 == 32 -> lane halves hit disjoint banks
#define ROWS_PER_WG 64              // 4 waves * 16 token-rows
#define THREADS   128
#define ASYNC_CALLS 20              // per thread per chunk: 4 x-b128 + 16 W-b32

#if __has_builtin(__builtin_amdgcn_global_load_async_to_lds_b128) && \
    __has_builtin(__builtin_amdgcn_global_load_async_to_lds_b32)
#define HAVE_ASYNC_LDS 1
#endif

// --- 16B global -> LDS copy: async (ASYNCcnt) if available, sync fallback ---
__device__ __forceinline__ void copy16_g2l(const float* __restrict__ g,
                                           float* __restrict__ l) {
#ifdef HAVE_ASYNC_LDS
  __builtin_amdgcn_global_load_async_to_lds_b128(
      (AS1 v4i*)g, (AS3 v4i*)l, /*offset=*/0, /*cpol=*/0);
#else
  float4 v = *(const float4*)g;
  l[0] = v.x; l[1] = v.y; l[2] = v.z; l[3] = v.w;
#endif
}

// --- 4B global -> LDS copy (scattered dest for interleaved W layout) ---
__device__ __forceinline__ void copy4_g2l(const float* __restrict__ g,
                                          float* __restrict__ l) {
#ifdef HAVE_ASYNC_LDS
  __builtin_amdgcn_global_load_async_to_lds_b32(
      (AS1 int*)g, (AS3 int*)l, /*offset=*/0, /*cpol=*/0);
#else
  *l = *g;
#endif
}

template <int N>
__device__ __forceinline__ void wait_async_le() {
#ifdef HAVE_ASYNC_LDS
#if __has_builtin(__builtin_amdgcn_s_wait_asynccnt)
  __builtin_amdgcn_s_wait_asynccnt(N);
#else
  asm volatile("s_wait_asynccnt %0" ::"i"(N) : "memory");
#endif
#endif
}

// ---------------------------------------------------------------------------
// Kernel 1: logits[t][e] = sum_k x[t][k] * W[k][e]
// One wave = 16 tokens x 64 experts (4 WMMA N-tiles, v8f accumulators).
// ---------------------------------------------------------------------------
__global__ __launch_bounds__(THREADS)
void gate_gemm_logits(const float* __restrict__ x,
                      const float* __restrict__ W,
                      float* __restrict__ logits) {
  __shared__ float xs[2][ROWS_PER_WG * XS_STRIDE];     // 2*64*36*4  = 18432 B
  __shared__ float ws[2][(BK / 2) * WP_STRIDE];        // 2*16*160*4 = 20480 B

  const int tid     = threadIdx.x;
  const int wave    = tid >> 5;
  const int lane    = tid & 31;
  const int half    = lane >> 4;    // 0: lanes 0-15, 1: lanes 16-31
  const int l16     = lane & 15;
  const int rowBase = blockIdx.x * ROWS_PER_WG;  // first global token row of WG
  const int waveRow = wave * 16;                 // first row within WG

  v8f acc[4];
#pragma unroll
  for (int t = 0; t < 4; ++t) {
    v8f z = {0.f, 0.f, 0.f, 0.f, 0.f, 0.f, 0.f, 0.f};
    acc[t] = z;
  }

  // --- cooperative chunk loader: x [64 x 32] and W [32 x 64] into buf ---
  // W goes in pair-interleaved: element (r, c) -> dword (r>>1)*160 + 2c + (r&1)
  auto load_chunk = [&](int kc, int buf) {
#pragma unroll
    for (int i = 0; i < 4; ++i) {          // x: 64 rows * 8 float4 = 512
      int idx = tid + i * THREADS;
      int r   = idx >> 3;
      int c4  = idx & 7;
      copy16_g2l(x + (size_t)(rowBase + r) * D_MODEL + kc + c4 * 4,
                 &xs[buf][r * XS_STRIDE + c4 * 4]);
    }
#pragma unroll
    for (int i = 0; i < 16; ++i) {         // W: 32 rows * 64 cols = 2048 scalars
      int idx = tid + i * THREADS;
      int r   = idx >> 6;                  // row in chunk (0..31)
      int c   = idx & 63;                  // expert column
      copy4_g2l(W + (size_t)(kc + r) * EXPERTS + c,
                &ws[buf][(r >> 1) * WP_STRIDE + 2 * c + (r & 1)]);
    }
  };

  // --- pipeline prologue: chunk 0 -> buffer 0 ---
  load_chunk(0, 0);

  int cur = 0;
  for (int kc = 0; kc < D_MODEL; kc += BK) {
    __syncthreads();  // all waves done reading buf[1-cur] (prev compute)

    if (kc + BK < D_MODEL) {
      load_chunk(kc + BK, 1 - cur);  // prefetch next chunk into other buffer
      wait_async_le<ASYNC_CALLS>();  // in-order: current chunk's copies landed
    } else {
      wait_async_le<0>();
    }
    __syncthreads();  // all waves' copies for chunk kc visible

    // --- compute: 8 K-steps of 4, 4 WMMA N-tiles each ---
#pragma unroll
    for (int k0 = 0; k0 < BK; k0 += 4) {
      // A fragment (16x4 f32): lanes 0-15 -> K k0..k0+1, lanes 16-31 -> k0+2..k0+3
      const float* ap = &xs[cur][(waveRow + l16) * XS_STRIDE + k0 + 2 * half];
      v2f a = *(const v2f*)ap;

      // B fragments: pair-row m holds K rows (2m, 2m+1) interleaved per column,
      // so {W[k][n], W[k+1][n]} is one aligned 8-byte ds_load_b64.
      const float* bp = &ws[cur][((k0 >> 1) + half) * WP_STRIDE + 2 * l16];
#pragma unroll
      for (int t = 0; t < 4; ++t) {
        v2f b = *(const v2f*)(bp + 32 * t);
        acc[t] = __builtin_amdgcn_wmma_f32_16x16x4_f32(
            /*neg_a=*/false, a, /*neg_b=*/false, b,
            /*c_mod=*/(short)0, acc[t],
            /*reuse_a=*/false, /*reuse_b=*/false);
      }
    }
    cur ^= 1;
  }

  // --- store logits: C layout = VGPR j holds row j (lanes 0-15) / j+8 (16-31) ---
#pragma unroll
  for (int t = 0; t < 4; ++t) {
#pragma unroll
    for (int j = 0; j < 8; ++j) {
      int row = rowBase + waveRow + j + 8 * half;
      int col = t * 16 + l16;
      logits[(size_t)row * EXPERTS + col] = acc[t][j];
    }
  }
}

// ---------------------------------------------------------------------------
// Kernel 2: softmax over 64 experts + top-2 mask. One thread per token.
// Tiny (8 MB, L2-hot after kernel 1); first-occurrence tie-break matches
// jax.lax.top_k.
// ---------------------------------------------------------------------------
__global__ __launch_bounds__(256)
void gate_softmax_top2(const float* __restrict__ logits,
                       float* __restrict__ y) {
  int t = blockIdx.x * blockDim.x + threadIdx.x;
  if (t >= TOKENS) return;
  const float* l = logits + (size_t)t * EXPERTS;

  float m = -INFINITY;
#pragma unroll
  for (int e = 0; e < EXPERTS; ++e) m = fmaxf(m, l[e]);

  float s = 0.f;
#pragma unroll
  for (int e = 0; e < EXPERTS; ++e) s += __expf(l[e] - m);
  float inv = 1.f / s;

  int i1 = -1; float v1 = -INFINITY;
#pragma unroll
  for (int e = 0; e < EXPERTS; ++e) {
    float v = l[e];
    if (v > v1) { v1 = v; i1 = e; }
  }
  int i2 = -1; float v2 = -INFINITY;
#pragma unroll
  for (int e = 0; e < EXPERTS; ++e) {
    if (e == i1) continue;
    float v = l[e];
    if (v > v2) { v2 = v; i2 = e; }
  }

  float* yo = y + (size_t)t * EXPERTS;
#pragma unroll
  for (int e = 0; e < EXPERTS; ++e) {
    float p = __expf(l[e] - m) * inv;
    yo[e] = (e == i1 || e == i2) ? p : 0.f;
  }
}

// ---------------------------------------------------------------------------
extern "C" void kernel_launch(void* const* d_in, const int* in_sizes, int n_in,
                              void* d_out, int out_size, void* d_ws,
                              size_t ws_size, hipStream_t stream) {
  const float* x = (const float*)d_in[0];                       // [4,4096,4096]
  const float* W = (const float*)d_in[1];                       // [4096,64]
  float* y      = (float*)d_out;                                // output 0
  float* logits = (float*)d_out + (size_t)TOKENS * EXPERTS;     // output 1

  dim3 g1(TOKENS / ROWS_PER_WG);   // 256 workgroups
  gate_gemm_logits<<<g1, THREADS, 0, stream>>>(x, W, logits);

  dim3 g2((TOKENS + 255) / 256);
  gate_softmax_top2<<<g2, 256, 0, stream>>>(logits, y);
}